// SnakeFastLoss_45088566674039
// MI455X (gfx1250) — compile-verified
//
#include <hip/hip_runtime.h>

#define HH 256
#define WW 256
#define BB 4
#define NN 64
#define HW (HH * WW)

typedef __attribute__((ext_vector_type(2))) float v2f;
typedef __attribute__((ext_vector_type(8))) float v8f;

// 13-tap Gaussian (std=2, normalized to sum 1) and derivative (normalized so |.|sum = 2)
__constant__ float G_TAP[13] = {
    0.0022182f, 0.0087731f, 0.0270232f, 0.0648252f, 0.1211094f, 0.1762132f,
    0.1996757f,
    0.1762132f, 0.1211094f, 0.0648252f, 0.0270232f, 0.0087731f, 0.0022182f};
__constant__ float DG_TAP[13] = {
    0.0171031f, 0.0563699f, 0.1389053f, 0.2499126f, 0.3112641f, 0.2264439f,
    0.0f,
   -0.2264439f,-0.3112641f,-0.2499126f,-0.1389053f,-0.0563699f,-0.0171031f};

// ---------------------------------------------------------------------------
// Build MINV = inv(I + 0.1*(0.01*D1'D1 + 0.01*D2'D2)) on device (Gauss-Jordan).
// ---------------------------------------------------------------------------
__global__ __launch_bounds__(64) void snake_build_minv(float* __restrict__ minv) {
  __shared__ float A[64][64];
  __shared__ float Inv[64][64];
  const int i = threadIdx.x;
  for (int j = 0; j < 64; ++j) {
    int d = i - j, ad = d < 0 ? -d : d;
    float l1 = 0.f, l2 = 0.f;
    if (ad == 0) {
      l1 = (i == 0 || i == 63) ? 1.f : 2.f;
      l2 = (i == 0 || i == 63) ? 1.f : (i == 1 || i == 62) ? 5.f : 6.f;
    } else if (ad == 1) {
      l1 = -1.f;
      int mn = i < j ? i : j, mx = i < j ? j : i;
      l2 = (mn == 0 || mx == 63) ? -2.f : -4.f;
    } else if (ad == 2) {
      l2 = 1.f;
    }
    A[i][j] = ((i == j) ? 1.f : 0.f) + 0.1f * (0.01f * l1 + 0.01f * l2);
    Inv[i][j] = (i == j) ? 1.f : 0.f;
  }
  __syncthreads();
  for (int p = 0; p < 64; ++p) {
    if (i == p) {
      float pv = 1.0f / A[p][p];
      for (int j = 0; j < 64; ++j) { A[p][j] *= pv; Inv[p][j] *= pv; }
    }
    __syncthreads();
    if (i != p) {
      float f = A[i][p];
      for (int j = 0; j < 64; ++j) { A[i][j] -= f * A[p][j]; Inv[i][j] -= f * Inv[p][j]; }
    }
    __syncthreads();
  }
  for (int j = 0; j < 64; ++j) minv[i * 64 + j] = Inv[i][j];
}

// ---------------------------------------------------------------------------
// Separable 13-tap conv, pass 1 (along w).
// ---------------------------------------------------------------------------
__global__ __launch_bounds__(256) void snake_conv_rows(const float* __restrict__ in,
                                                       float* __restrict__ tmp) {
  const int bh = blockIdx.x;
  const int b = bh >> 8, h = bh & 255;
  const int w = threadIdx.x;
  __shared__ float row[WW];
  row[w] = in[b * HW + h * WW + w];
  __syncthreads();
  float sg = 0.f, sd = 0.f;
#pragma unroll
  for (int j = 0; j < 13; ++j) {
    int ww = w + j - 6;
    float v = (ww >= 0 && ww < WW) ? row[ww] : 0.0f;
    sg = fmaf(v, G_TAP[j], sg);
    sd = fmaf(v, DG_TAP[j], sd);
  }
  tmp[(b * 2 + 0) * HW + h * WW + w] = sg;
  tmp[(b * 2 + 1) * HW + h * WW + w] = sd;
}

// Pass 2 (along h): ch0 = dg over sg, ch1 = g over sd, x EXTGRADFAC(=10).
__global__ __launch_bounds__(256) void snake_conv_cols(const float* __restrict__ tmp,
                                                       float* __restrict__ gimg) {
  const int bh = blockIdx.x;
  const int b = bh >> 8, h = bh & 255;
  const int w = threadIdx.x;
  float o0 = 0.f, o1 = 0.f;
#pragma unroll
  for (int i = 0; i < 13; ++i) {
    int hh = h + i - 6;
    if (hh >= 0 && hh < HH) {
      float vg = tmp[(b * 2 + 0) * HW + hh * WW + w];
      float vd = tmp[(b * 2 + 1) * HW + hh * WW + w];
      o0 = fmaf(vg, DG_TAP[i], o0);
      o1 = fmaf(vd, G_TAP[i], o1);
    }
  }
  gimg[(b * 2 + 0) * HW + h * WW + w] = o0 * 10.0f;
  gimg[(b * 2 + 1) * HW + h * WW + w] = o1 * 10.0f;
}

// ---------------------------------------------------------------------------
// Snake relaxation: one wave32 per batch; 32 sequential steps of
//   pos = clip(MINV @ (pos + 0.1*bilinear(gimg,pos)))
// via V_WMMA_F32_16X16X4_F32. A (MINV) lives in registers (4x16 v2f frags),
// B (padded RHS) in LDS in exact fragment layout -> one ds_load_b64 per K
// step, shared by all 4 M-tiles; 4 interleaved accumulator chains.
// ---------------------------------------------------------------------------
__device__ __forceinline__ float snake_bilin(const float* __restrict__ g,
                                             float r, float c) {
  r = fminf(fmaxf(r, 0.f), 255.f);
  c = fminf(fmaxf(c, 0.f), 255.f);
  float r0 = fminf(floorf(r), 254.f);
  float c0 = fminf(floorf(c), 254.f);
  float wr = r - r0, wc = c - c0;
  int r0i = (int)r0, c0i = (int)c0;
  const float* p = g + r0i * WW + c0i;
  float g00 = p[0], g01 = p[1], g10 = p[WW], g11 = p[WW + 1];
  return g00 * (1.f - wr) * (1.f - wc) + g01 * (1.f - wr) * wc +
         g10 * wr * (1.f - wc) + g11 * wr * wc;
}

#define WMMA_F32(a, b, c) \
  __builtin_amdgcn_wmma_f32_16x16x4_f32(false, (a), false, (b), (short)0, (c), false, false)

__global__ __launch_bounds__(32) void snake_optim_wmma(const float* __restrict__ gimg,
                                                       const float* __restrict__ minv,
                                                       const float* __restrict__ pos0,
                                                       float* __restrict__ posOut) {
  // B operand in fragment layout: sxB[col*64 + k]; cols 2..15 stay zero.
  __shared__ float sxB[16 * 64];
  __shared__ float bufR[64], bufC[64];

  const int lane = threadIdx.x;
  const int b = blockIdx.x;
  const float* g0 = gimg + (b * 2 + 0) * HW;
  const float* g1 = gimg + (b * 2 + 1) * HW;

  const int col = lane & 15;           // N index (B/D column)
  const int khalf = (lane >> 4) << 1;  // lanes 16-31 carry K+2/K+3
  const int mloc = lane & 15;          // M row within a 16-row tile (A frag)

  // A fragments: MINV resident in registers for the whole 32-step loop.
  v2f aReg[4][16];
#pragma unroll
  for (int mt = 0; mt < 4; ++mt) {
#pragma unroll
    for (int kt = 0; kt < 16; ++kt) {
      aReg[mt][kt] = *(const v2f*)&minv[(mt * 16 + mloc) * 64 + kt * 4 + khalf];
    }
  }

  // zero the padded B columns once; load initial positions
  for (int i = lane + 128; i < 1024; i += 32) sxB[i] = 0.0f;
  for (int n = lane; n < 64; n += 32) {
    bufR[n] = pos0[b * 128 + n * 2 + 0];
    bufC[n] = pos0[b * 128 + n * 2 + 1];
  }
  __syncthreads();

  for (int s = 0; s < 32; ++s) {
    // x = pos + 0.1 * f(pos); write into B-fragment columns 0 (r) and 1 (c)
    for (int n = lane; n < 64; n += 32) {
      float r = bufR[n], c = bufC[n];
      sxB[0 * 64 + n] = r + 0.1f * snake_bilin(g0, r, c);
      sxB[1 * 64 + n] = c + 0.1f * snake_bilin(g1, r, c);
    }
    __syncthreads();

    v8f acc0 = {}, acc1 = {}, acc2 = {}, acc3 = {};
#pragma unroll
    for (int kt = 0; kt < 16; ++kt) {
      v2f bb = *(const v2f*)&sxB[col * 64 + kt * 4 + khalf];
      acc0 = WMMA_F32(aReg[0][kt], bb, acc0);
      acc1 = WMMA_F32(aReg[1][kt], bb, acc1);
      acc2 = WMMA_F32(aReg[2][kt], bb, acc2);
      acc3 = WMMA_F32(aReg[3][kt], bb, acc3);
    }
    __syncthreads();  // all B-frag reads done before pos rewrite

    // D layout: VGPR v -> row (v | v+8) of the tile, col = lane&15.
    if (col < 2) {
      const int rowBase = (lane >> 4) << 3;
      float* dst = (col == 0) ? bufR : bufC;
#pragma unroll
      for (int v = 0; v < 8; ++v) {
        dst[ 0 + rowBase + v] = fminf(fmaxf(acc0[v], 0.0f), 255.0f);
        dst[16 + rowBase + v] = fminf(fmaxf(acc1[v], 0.0f), 255.0f);
        dst[32 + rowBase + v] = fminf(fmaxf(acc2[v], 0.0f), 255.0f);
        dst[48 + rowBase + v] = fminf(fmaxf(acc3[v], 0.0f), 255.0f);
      }
    }
    __syncthreads();
  }

  for (int n = lane; n < 64; n += 32) {
    posOut[b * 128 + n * 2 + 0] = bufR[n];
    posOut[b * 128 + n * 2 + 1] = bufC[n];
  }
}

// ---------------------------------------------------------------------------
// Distance-map render + MSE partial sums (deterministic two-stage reduce).
// ---------------------------------------------------------------------------
__global__ __launch_bounds__(256) void snake_render_loss(const float* __restrict__ pred,
                                                         const float* __restrict__ pos,
                                                         float* __restrict__ partials) {
  const int b = blockIdx.y;
  const int t = threadIdx.x;
  __shared__ float sa0[63], sa1[63], sab0[63], sab1[63], sinv[63];
  if (t < 63) {
    float a0 = pos[b * 128 + t * 2 + 0], a1 = pos[b * 128 + t * 2 + 1];
    float b0 = pos[b * 128 + t * 2 + 2], b1 = pos[b * 128 + t * 2 + 3];
    float ab0 = b0 - a0, ab1 = b1 - a1;
    sa0[t] = a0; sa1[t] = a1; sab0[t] = ab0; sab1[t] = ab1;
    sinv[t] = 1.0f / (ab0 * ab0 + ab1 * ab1 + 1e-8f);
  }
  __syncthreads();
  const int idx = blockIdx.x * 256 + t;
  const float ph = (float)(idx >> 8);   // row (H index)
  const float pw = (float)(idx & 255);  // col (W index)
  float best = 3.4e38f;
#pragma unroll 7
  for (int s = 0; s < 63; ++s) {
    float ap0 = ph - sa0[s], ap1 = pw - sa1[s];
    float tp = (ap0 * sab0[s] + ap1 * sab1[s]) * sinv[s];
    tp = fminf(fmaxf(tp, 0.f), 1.f);
    float d0 = ap0 - tp * sab0[s];
    float d1 = ap1 - tp * sab1[s];
    best = fminf(best, d0 * d0 + d1 * d1);
  }
  float d = fminf(sqrtf(best + 1e-12f), 15.0f);
  float diff = pred[b * HW + idx] - d;
  __shared__ float red[256];
  red[t] = diff * diff;
  __syncthreads();
  for (int off = 128; off > 0; off >>= 1) {
    if (t < off) red[t] += red[t + off];
    __syncthreads();
  }
  if (t == 0) partials[b * gridDim.x + blockIdx.x] = red[0];
}

__global__ __launch_bounds__(256) void snake_finalize(const float* __restrict__ partials,
                                                      float* __restrict__ out,
                                                      int n, float scale) {
  __shared__ float red[256];
  float s = 0.f;
  for (int i = threadIdx.x; i < n; i += 256) s += partials[i];
  red[threadIdx.x] = s;
  __syncthreads();
  for (int off = 128; off > 0; off >>= 1) {
    if (threadIdx.x < off) red[threadIdx.x] += red[threadIdx.x + off];
    __syncthreads();
  }
  if (threadIdx.x == 0) out[0] = red[0] * scale;
}

// ---------------------------------------------------------------------------
extern "C" void kernel_launch(void* const* d_in, const int* in_sizes, int n_in,
                              void* d_out, int out_size, void* d_ws, size_t ws_size,
                              hipStream_t stream) {
  const float* pred = (const float*)d_in[0];   // (B,1,H,W) f32
  const float* pos0 = (const float*)d_in[1];   // (B,N,2) f32
  float* ws = (float*)d_ws;

  float* tmp      = ws;                         // B*2*H*W = 524288 floats
  float* gimg     = ws + 524288;                // B*2*H*W = 524288 floats
  float* minv     = ws + 1048576;               // 4096 floats
  float* pos      = ws + 1052672;               // 512 floats
  float* partials = ws + 1053184;               // 1024 floats

  snake_build_minv<<<1, 64, 0, stream>>>(minv);
  snake_conv_rows<<<BB * HH, 256, 0, stream>>>(pred, tmp);
  snake_conv_cols<<<BB * HH, 256, 0, stream>>>(tmp, gimg);
  snake_optim_wmma<<<BB, 32, 0, stream>>>(gimg, minv, pos0, pos);
  snake_render_loss<<<dim3(HW / 256, BB), 256, 0, stream>>>(pred, pos, partials);
  snake_finalize<<<1, 256, 0, stream>>>(partials, (float*)d_out, BB * (HW / 256),
                                        1.0f / (float)(BB * HW));
}